// DNCCell_10075993276872
// MI455X (gfx1250) — compile-verified
//
#include <hip/hip_runtime.h>
#include <math.h>

// Problem constants (from reference setup_inputs)
#define BB     16
#define I_DIM  256
#define H_DIM  512
#define O_DIM  256
#define A_DIM  1024
#define M_DIM  64
#define R_DIM  4
#define IF_DIM 471
#define Z_DIM  2048   // 4*H
#define XR_DIM 512    // I + R*M

typedef float v2f __attribute__((ext_vector_type(2)));
typedef float v8f __attribute__((ext_vector_type(8)));

// ---------- workspace layout (float offsets) ----------
#define WS_XR     0u                         // 16*512
#define WS_Z      (WS_XR   + BB*XR_DIM)      // 16*2048
#define WS_H      (WS_Z    + BB*Z_DIM)       // 16*512
#define WS_OUTH   (WS_H    + BB*H_DIM)       // 16*256
#define WS_IFACE  (WS_OUTH + BB*O_DIM)       // 16*471
#define WS_BETAR  (WS_IFACE+ BB*IF_DIM)      // 16*4
#define WS_BETAW  (WS_BETAR+ BB*R_DIM)       // 16
#define WS_GA     (WS_BETAW+ BB)             // 16
#define WS_GW     (WS_GA   + BB)             // 16
#define WS_FREE   (WS_GW   + BB)             // 16*4
#define WS_PI     (WS_FREE + BB*R_DIM)       // 16*4*3
#define WS_ERASE  (WS_PI   + BB*R_DIM*3)     // 16*64
#define WS_ALLOC  (WS_ERASE+ BB*M_DIM)       // 16*1024
#define WS_CW     (WS_ALLOC+ BB*A_DIM)       // 16*1024
#define WS_WW     (WS_CW   + BB*A_DIM)       // 16*1024
#define WS_PREC   (WS_WW   + BB*A_DIM)       // 16*1024
#define WS_MEM    (WS_PREC + BB*A_DIM)       // 16*1024*64
#define WS_FWD    (WS_MEM  + BB*A_DIM*M_DIM) // 16*4*1024
#define WS_BWD    (WS_FWD  + BB*R_DIM*A_DIM) // 16*4*1024
#define WS_CR     (WS_BWD  + BB*R_DIM*A_DIM) // 16*4*1024
#define WS_RV     (WS_CR   + BB*R_DIM*A_DIM) // 16*256

// ---------- device math helpers ----------
__device__ __forceinline__ float dsigmoid(float x) { return 1.f / (1.f + expf(-x)); }
__device__ __forceinline__ float doneplus(float x) {
    return 1.f + ((x > 20.f) ? x : log1pf(expf(x)));
}

// ---------- WMMA 16x16 fp32 tile helpers (V_WMMA_F32_16X16X4_F32) ----------
// Branch-free inner loop: out-of-range columns are clamped to N-1 (duplicate,
// always in-bounds loads); their results are discarded by the guarded store.
// Pointer-increment addressing lets the compiler strength-reduce all index math.
__device__ __forceinline__ void wmma_loop(const float* __restrict__ Amat, int lda,
                                          const float* __restrict__ Wmat, int ldw,
                                          int K, int n0, int N, int lane, v8f& acc)
{
    const int m  = lane & 15;
    const int kh = lane >> 4;      // 0 or 1
    int col = n0 + (lane & 15);
    if (col >= N) col = N - 1;     // clamp (results never stored)
    // A 16x4: lanes 0-15 -> K={k,k+1}, lanes 16-31 -> K={k+2,k+3}
    const float* ap = Amat + (size_t)m * lda + 2 * kh;
    // B 4x16: row striped across 16 lanes; halves hold K rows {0,1} / {2,3}
    const float* wp = Wmat + (size_t)(2 * kh) * ldw + col;
    const size_t wstep = (size_t)4 * ldw;
#pragma unroll 4
    for (int k = 0; k < K; k += 4) {
        v2f a, w;
        a.x = ap[0];               // contiguous pair -> global_load_b64
        a.y = ap[1];
        w.x = wp[0];
        w.y = wp[ldw];
        ap += 4;
        wp += wstep;
        acc = __builtin_amdgcn_wmma_f32_16x16x4_f32(false, a, false, w,
                                                    (short)0, acc, false, false);
    }
}

__device__ __forceinline__ void wmma_store(float* __restrict__ Out, int ldo,
                                           const float* __restrict__ bias,
                                           const float* __restrict__ add,
                                           int n0, int N, int lane, const v8f& acc)
{
    const int kh  = lane >> 4;
    const int col = n0 + (lane & 15);
    if (col >= N) return;
    const float bv = bias ? bias[col] : 0.f;
#pragma unroll
    for (int v = 0; v < 8; v++) {
        const int row = v + 8 * kh;   // C/D layout: VGPR v -> M=v (+8 for upper half)
        float o = acc[v] + bv;
        if (add) o += add[(size_t)row * ldo + col];
        Out[(size_t)row * ldo + col] = o;
    }
}

// C[16,N] = A[16,K] @ W[K,N] + bias (+ add)
__global__ void gemm16_kernel(const float* __restrict__ A, int lda,
                              const float* __restrict__ W,
                              const float* __restrict__ bias,
                              const float* __restrict__ add,
                              float* __restrict__ Out, int K, int N)
{
    const int wid  = (blockIdx.x * blockDim.x + threadIdx.x) >> 5;
    const int lane = threadIdx.x & 31;
    const int n0   = wid * 16;
    if (n0 >= N) return;               // wave-uniform exit (EXEC stays all-ones for WMMA)
    v8f acc = {};
    wmma_loop(A, lda, W, N, K, n0, N, lane, acc);
    wmma_store(Out, N, bias, add, n0, N, lane, acc);
}

// C[16,N] = A1@W1 + A2@W2 + bias   (LSTM pre-activations)
__global__ void gemm16_dual_kernel(const float* __restrict__ A1, const float* __restrict__ W1,
                                   const float* __restrict__ A2, const float* __restrict__ W2,
                                   const float* __restrict__ bias,
                                   float* __restrict__ Out, int K, int N)
{
    const int wid  = (blockIdx.x * blockDim.x + threadIdx.x) >> 5;
    const int lane = threadIdx.x & 31;
    const int n0   = wid * 16;
    if (n0 >= N) return;
    v8f acc = {};
    wmma_loop(A1, K, W1, N, K, n0, N, lane, acc);
    wmma_loop(A2, K, W2, N, K, n0, N, lane, acc);
    wmma_store(Out, N, bias, nullptr, n0, N, lane, acc);
}

// ---------- small elementwise kernels ----------
__global__ void concat_kernel(const float* __restrict__ x, const float* __restrict__ rv_prev,
                              float* __restrict__ xr)
{
    int i = blockIdx.x * 256 + threadIdx.x;        // < 16*512
    int b = i / XR_DIM, t = i % XR_DIM;
    xr[i] = (t < I_DIM) ? x[b * I_DIM + t] : rv_prev[b * I_DIM + (t - I_DIM)];
}

__global__ void lstm_kernel(const float* __restrict__ z, const float* __restrict__ c_prev,
                            float* __restrict__ h)
{
    int i = blockIdx.x * 256 + threadIdx.x;        // < 16*512
    int b = i / H_DIM, hh = i % H_DIM;
    const float* zb = z + (size_t)b * Z_DIM;
    float iv = dsigmoid(zb[hh]);
    float fv = dsigmoid(zb[H_DIM + hh]);
    float gv = tanhf(zb[2 * H_DIM + hh]);
    float ov = dsigmoid(zb[3 * H_DIM + hh]);
    float cv = fv * c_prev[i] + iv * gv;
    h[i] = ov * tanhf(cv);
}

__global__ void iface_kernel(const float* __restrict__ iface, float* __restrict__ ws)
{
    int b = blockIdx.x, tid = threadIdx.x;
    const float* f = iface + (size_t)b * IF_DIM;
    if (tid < 4)            ws[WS_BETAR + b * 4 + tid] = doneplus(f[256 + tid]);
    else if (tid == 4)      ws[WS_BETAW + b]           = doneplus(f[324]);
    else if (tid == 5)      ws[WS_GA + b]              = dsigmoid(f[457]);
    else if (tid == 6)      ws[WS_GW + b]              = dsigmoid(f[458]);
    else if (tid >= 8 && tid < 12)
        ws[WS_FREE + b * 4 + (tid - 8)] = dsigmoid(f[453 + (tid - 8)]);
    else if (tid >= 16 && tid < 20) {
        int r = tid - 16;
        float v0 = f[459 + 3 * r], v1 = f[460 + 3 * r], v2 = f[461 + 3 * r];
        float mx = fmaxf(v0, fmaxf(v1, v2));
        float e0 = expf(v0 - mx), e1 = expf(v1 - mx), e2 = expf(v2 - mx);
        float inv = 1.f / (e0 + e1 + e2);
        ws[WS_PI + (b * 4 + r) * 3 + 0] = e0 * inv;
        ws[WS_PI + (b * 4 + r) * 3 + 1] = e1 * inv;
        ws[WS_PI + (b * 4 + r) * 3 + 2] = e2 * inv;
    } else if (tid >= 64 && tid < 128)
        ws[WS_ERASE + b * 64 + (tid - 64)] = dsigmoid(f[325 + (tid - 64)]);
}

// ---------- usage + stable argsort + allocation (one block / batch) ----------
__global__ void alloc_kernel(const float* __restrict__ usage_prev,
                             const float* __restrict__ ww_prev,
                             const float* __restrict__ rw_prev,
                             const float* __restrict__ free_, float* __restrict__ alloc_out)
{
    __shared__ float val[A_DIM];
    __shared__ int   idx[A_DIM];
    int b = blockIdx.x, tid = threadIdx.x;       // 512 threads
    for (int a = tid; a < A_DIM; a += 512) {
        float psi = 1.f;
#pragma unroll
        for (int r = 0; r < R_DIM; r++)
            psi *= (1.f - free_[b * R_DIM + r] * rw_prev[((size_t)b * R_DIM + r) * A_DIM + a]);
        float u = usage_prev[b * A_DIM + a], w = ww_prev[b * A_DIM + a];
        val[a] = (u + w - u * w) * psi;
        idx[a] = a;
    }
    __syncthreads();
    // bitonic ascending, index tiebreak => matches stable jnp.argsort
    for (int k = 2; k <= A_DIM; k <<= 1) {
        for (int j = k >> 1; j > 0; j >>= 1) {
            for (int t = tid; t < A_DIM; t += 512) {
                int p = t ^ j;
                if (p > t) {
                    bool up = ((t & k) == 0);
                    float v1 = val[t], v2 = val[p];
                    int   i1 = idx[t], i2 = idx[p];
                    bool gt = (v1 > v2) || (v1 == v2 && i1 > i2);
                    if (gt == up) { val[t] = v2; val[p] = v1; idx[t] = i2; idx[p] = i1; }
                }
            }
            __syncthreads();
        }
    }
    if (tid == 0) {
        float cp = 1.f;
        for (int t = 0; t < A_DIM; t++) {
            float su = val[t];
            alloc_out[b * A_DIM + idx[t]] = (1.f - su) * cp;   // scatter back
            cp *= su;
        }
    }
}

// ---------- content addressing: softmax_a(beta * cos(key, mem[a])) ----------
__global__ void content_kernel(const float* __restrict__ mem, const float* __restrict__ keys,
                               int keyBStride, const float* __restrict__ beta, int nkeys,
                               float* __restrict__ out)
{
    __shared__ float kL[M_DIM];
    __shared__ float red[256];
    int blk = blockIdx.x, b = blk / nkeys, r = blk % nkeys, tid = threadIdx.x;
    const float* key = keys + (size_t)b * keyBStride + r * M_DIM;
    if (tid < M_DIM) kL[tid] = key[tid];
    __syncthreads();
    float kn = 0.f;
#pragma unroll
    for (int m = 0; m < M_DIM; m++) kn += kL[m] * kL[m];
    kn = sqrtf(kn);
    const float bet = beta[b * nkeys + r];
    const float* Mb = mem + (size_t)b * A_DIM * M_DIM;
    float v[4];
#pragma unroll
    for (int p = 0; p < 4; p++) {
        int a = tid + p * 256;
        const float* row = Mb + (size_t)a * M_DIM;
        float dot = 0.f, nn = 0.f;
        for (int m = 0; m < M_DIM; m++) { float e = row[m]; dot += e * kL[m]; nn += e * e; }
        v[p] = bet * (dot / (kn * sqrtf(nn) + 1e-6f));
    }
    float mx = fmaxf(fmaxf(v[0], v[1]), fmaxf(v[2], v[3]));
    red[tid] = mx; __syncthreads();
    for (int s = 128; s > 0; s >>= 1) { if (tid < s) red[tid] = fmaxf(red[tid], red[tid + s]); __syncthreads(); }
    float gmax = red[0]; __syncthreads();
    float sum = 0.f;
#pragma unroll
    for (int p = 0; p < 4; p++) { v[p] = expf(v[p] - gmax); sum += v[p]; }
    red[tid] = sum; __syncthreads();
    for (int s = 128; s > 0; s >>= 1) { if (tid < s) red[tid] += red[tid + s]; __syncthreads(); }
    float inv = 1.f / red[0];
#pragma unroll
    for (int p = 0; p < 4; p++)
        out[((size_t)b * nkeys + r) * A_DIM + tid + p * 256] = v[p] * inv;
}

// ---------- ww, sum(ww), prec ----------
__global__ void ww_prec_kernel(const float* __restrict__ alloc_, const float* __restrict__ cw,
                               const float* __restrict__ ga_, const float* __restrict__ gw_,
                               const float* __restrict__ prec_prev,
                               float* __restrict__ ww, float* __restrict__ prec)
{
    __shared__ float red[256];
    int b = blockIdx.x, tid = threadIdx.x;
    float ga = ga_[b], gw = gw_[b], part = 0.f, wl[4];
#pragma unroll
    for (int p = 0; p < 4; p++) {
        int a = tid + p * 256;
        float w = gw * (ga * alloc_[b * A_DIM + a] + (1.f - ga) * cw[b * A_DIM + a]);
        wl[p] = w; ww[b * A_DIM + a] = w; part += w;
    }
    red[tid] = part; __syncthreads();
    for (int s = 128; s > 0; s >>= 1) { if (tid < s) red[tid] += red[tid + s]; __syncthreads(); }
    float sum = red[0];
#pragma unroll
    for (int p = 0; p < 4; p++) {
        int a = tid + p * 256;
        prec[b * A_DIM + a] = (1.f - sum) * prec_prev[b * A_DIM + a] + wl[p];
    }
}

// ---------- memory write ----------
__global__ void mem_kernel(const float* __restrict__ mem_prev, const float* __restrict__ ww,
                           const float* __restrict__ erase, const float* __restrict__ iface,
                           float* __restrict__ mem)
{
    int i = blockIdx.x * 256 + threadIdx.x;        // < 16*1024*64
    int m = i & 63, a = (i >> 6) & 1023, b = i >> 16;
    float w  = ww[b * A_DIM + a];
    float e  = erase[b * M_DIM + m];
    float wv = iface[(size_t)b * IF_DIM + 389 + m];  // write_v, raw
    mem[i] = mem_prev[i] * (1.f - w * e) + w * wv;
}

__global__ void zero_kernel(float* __restrict__ p, int n)
{
    int i = blockIdx.x * 256 + threadIdx.x;
    if (i < n) p[i] = 0.f;
}

// ---------- fused link + fwd/bwd: single streaming pass over link_prev ----------
// link[i,j] = (1-ww[i]-ww[j])*link_prev[i,j] + ww[i]*prec_prev[j], diag=0 (never stored)
// fwd[r,i] = sum_j link[i,j]*rw[r,j] ; bwd[r,j] += sum_i link[i,j]*rw[r,i]
__global__ void linkfb_kernel(const float* __restrict__ link_prev,
                              const float* __restrict__ rw_prev,
                              const float* __restrict__ ww,
                              const float* __restrict__ prec_prev,
                              float* __restrict__ fwd, float* __restrict__ bwd)
{
    __shared__ float wwL[A_DIM];
    __shared__ float precL[A_DIM];
    __shared__ float rwL[R_DIM][A_DIM];
    __shared__ float bwdL[R_DIM][A_DIM];
    __shared__ float fwdL[64][R_DIM];
    const int blk = blockIdx.x, b = blk >> 4, tile = blk & 15;
    const int i0 = tile * 64;
    const int tid = threadIdx.x, lane = tid & 31;
    for (int a = tid; a < A_DIM; a += 256) {
        wwL[a]   = ww[b * A_DIM + a];
        precL[a] = prec_prev[b * A_DIM + a];
#pragma unroll
        for (int r = 0; r < R_DIM; r++) {
            rwL[r][a]  = rw_prev[((size_t)b * R_DIM + r) * A_DIM + a];
            bwdL[r][a] = 0.f;
        }
    }
    for (int t = tid; t < 64 * R_DIM; t += 256) fwdL[t >> 2][t & 3] = 0.f;
    __syncthreads();
    const float* lbase = link_prev + ((size_t)b * A_DIM + i0) * A_DIM;
    for (int ii = 0; ii < 64; ii++) {
        const int i = i0 + ii;
        const float wwi = wwL[i];
        const float r0 = rwL[0][i], r1 = rwL[1][i], r2 = rwL[2][i], r3 = rwL[3][i];
        float f0 = 0.f, f1 = 0.f, f2 = 0.f, f3 = 0.f;
        const float* lrow = lbase + (size_t)ii * A_DIM;
        if (ii + 1 < 64) __builtin_prefetch(lrow + A_DIM + tid * 4, 0, 0);
        for (int j = tid; j < A_DIM; j += 256) {
            float lv = lrow[j];
            float lw = (1.f - wwi - wwL[j]) * lv + wwi * precL[j];
            lw = (j == i) ? 0.f : lw;
            f0 += lw * rwL[0][j]; f1 += lw * rwL[1][j];
            f2 += lw * rwL[2][j]; f3 += lw * rwL[3][j];
            // slots j with j%256==tid are thread-owned -> race-free RMW
            bwdL[0][j] += lw * r0; bwdL[1][j] += lw * r1;
            bwdL[2][j] += lw * r2; bwdL[3][j] += lw * r3;
        }
#pragma unroll
        for (int off = 16; off > 0; off >>= 1) {          // wave32 reduction
            f0 += __shfl_xor(f0, off, 32);
            f1 += __shfl_xor(f1, off, 32);
            f2 += __shfl_xor(f2, off, 32);
            f3 += __shfl_xor(f3, off, 32);
        }
        if (lane == 0) {                                   // LDS float atomics (ds_add_f32)
            atomicAdd(&fwdL[ii][0], f0); atomicAdd(&fwdL[ii][1], f1);
            atomicAdd(&fwdL[ii][2], f2); atomicAdd(&fwdL[ii][3], f3);
        }
    }
    __syncthreads();
    for (int t = tid; t < 64 * R_DIM; t += 256) {
        int ii = t >> 2, r = t & 3;
        fwd[((size_t)b * R_DIM + r) * A_DIM + i0 + ii] = fwdL[ii][r];
    }
    for (int a = tid; a < A_DIM; a += 256)
#pragma unroll
        for (int r = 0; r < R_DIM; r++)
            atomicAdd(&bwd[((size_t)b * R_DIM + r) * A_DIM + a], bwdL[r][a]);
}

// ---------- w_r = pi0*bwd + pi1*c_r + pi2*fwd ; rv = w_r @ mem ----------
__global__ void wr_rv_kernel(const float* __restrict__ bwd, const float* __restrict__ cr,
                             const float* __restrict__ fwd, const float* __restrict__ piw,
                             const float* __restrict__ mem, float* __restrict__ rv)
{
    __shared__ float wL[A_DIM];
    __shared__ float part[4][M_DIM];
    const int blk = blockIdx.x, b = blk >> 2, r = blk & 3, tid = threadIdx.x;
    const float p0 = piw[(b * 4 + r) * 3 + 0];
    const float p1 = piw[(b * 4 + r) * 3 + 1];
    const float p2 = piw[(b * 4 + r) * 3 + 2];
    const int base = (b * 4 + r) * A_DIM;
    for (int a = tid; a < A_DIM; a += 256)
        wL[a] = p0 * bwd[base + a] + p1 * cr[base + a] + p2 * fwd[base + a];
    __syncthreads();
    const int m = tid & 63, p = tid >> 6;
    float acc = 0.f;
    const float* Mb = mem + (size_t)b * A_DIM * M_DIM;
    for (int a = p * 256; a < p * 256 + 256; a++)
        acc += wL[a] * Mb[(size_t)a * M_DIM + m];
    part[p][m] = acc;
    __syncthreads();
    if (tid < M_DIM)
        rv[(b * 4 + r) * M_DIM + tid] = part[0][tid] + part[1][tid] + part[2][tid] + part[3][tid];
}

// ============================= host launcher =============================
extern "C" void kernel_launch(void* const* d_in, const int* in_sizes, int n_in,
                              void* d_out, int out_size, void* d_ws, size_t ws_size,
                              hipStream_t stream)
{
    const float* x          = (const float*)d_in[0];
    const float* h_prev     = (const float*)d_in[1];
    const float* c_prev     = (const float*)d_in[2];
    const float* mem_prev   = (const float*)d_in[3];
    const float* rw_prev    = (const float*)d_in[4];
    const float* ww_prev    = (const float*)d_in[5];
    const float* usage_prev = (const float*)d_in[6];
    const float* prec_prev  = (const float*)d_in[7];
    const float* link_prev  = (const float*)d_in[8];
    const float* rv_prev    = (const float*)d_in[9];
    const float* Wx         = (const float*)d_in[10];
    const float* Wh         = (const float*)d_in[11];
    const float* b_lstm     = (const float*)d_in[12];
    const float* W_hid      = (const float*)d_in[13];
    const float* b_hid      = (const float*)d_in[14];
    const float* W_if       = (const float*)d_in[15];
    const float* b_if       = (const float*)d_in[16];
    const float* W_rd       = (const float*)d_in[17];
    const float* b_rd       = (const float*)d_in[18];
    float* out = (float*)d_out;
    float* ws  = (float*)d_ws;

    // 1) xr = concat(x, rv_prev)
    concat_kernel<<<(BB * XR_DIM) / 256, 256, 0, stream>>>(x, rv_prev, ws + WS_XR);
    // 2) z = xr@Wx + h_prev@Wh + b_lstm     (WMMA f32, N=2048 -> 128 waves)
    gemm16_dual_kernel<<<(Z_DIM / 16 + 7) / 8, 256, 0, stream>>>(
        ws + WS_XR, Wx, h_prev, Wh, b_lstm, ws + WS_Z, H_DIM, Z_DIM);
    // 3) h
    lstm_kernel<<<(BB * H_DIM) / 256, 256, 0, stream>>>(ws + WS_Z, c_prev, ws + WS_H);
    // 4) out_hidden = h@W_hid + b_hid
    gemm16_kernel<<<(O_DIM / 16 + 7) / 8, 256, 0, stream>>>(
        ws + WS_H, H_DIM, W_hid, b_hid, nullptr, ws + WS_OUTH, H_DIM, O_DIM);
    // 5) iface = h@W_if + b_if   (N=471, tail clamped in-loop, guarded store)
    gemm16_kernel<<<((IF_DIM + 15) / 16 + 7) / 8, 256, 0, stream>>>(
        ws + WS_H, H_DIM, W_if, b_if, nullptr, ws + WS_IFACE, H_DIM, IF_DIM);
    // 6) interface nonlinearities
    iface_kernel<<<BB, 256, 0, stream>>>(ws + WS_IFACE, ws);
    // 7) usage -> stable sort -> allocation weights
    alloc_kernel<<<BB, 512, 0, stream>>>(usage_prev, ww_prev, rw_prev,
                                         ws + WS_FREE, ws + WS_ALLOC);
    // 8) c_w on mem_prev (keys = iface[:,260:324], stride IF)
    content_kernel<<<BB, 256, 0, stream>>>(mem_prev, ws + WS_IFACE + 260, IF_DIM,
                                           ws + WS_BETAW, 1, ws + WS_CW);
    // 9) ww + prec
    ww_prec_kernel<<<BB, 256, 0, stream>>>(ws + WS_ALLOC, ws + WS_CW, ws + WS_GA,
                                           ws + WS_GW, prec_prev, ws + WS_WW, ws + WS_PREC);
    // 10) mem update
    mem_kernel<<<(BB * A_DIM * M_DIM) / 256, 256, 0, stream>>>(
        mem_prev, ws + WS_WW, ws + WS_ERASE, ws + WS_IFACE, ws + WS_MEM);
    // 11) zero bwd accumulator, 12) fused link stream -> fwd/bwd
    zero_kernel<<<(BB * R_DIM * A_DIM) / 256, 256, 0, stream>>>(ws + WS_BWD, BB * R_DIM * A_DIM);
    linkfb_kernel<<<BB * 16, 256, 0, stream>>>(link_prev, rw_prev, ws + WS_WW,
                                               prec_prev, ws + WS_FWD, ws + WS_BWD);
    // 13) c_r on new mem (keys = iface[:,0:256] as (R,M), stride IF)
    content_kernel<<<BB * R_DIM, 256, 0, stream>>>(ws + WS_MEM, ws + WS_IFACE, IF_DIM,
                                                   ws + WS_BETAR, R_DIM, ws + WS_CR);
    // 14) w_r mix + rv = w_r @ mem
    wr_rv_kernel<<<BB * R_DIM, 256, 0, stream>>>(ws + WS_BWD, ws + WS_CR, ws + WS_FWD,
                                                 ws + WS_PI, ws + WS_MEM, ws + WS_RV);
    // 15) out = out_hidden + rv@W_rd + b_rd   (WMMA)
    gemm16_kernel<<<(O_DIM / 16 + 7) / 8, 256, 0, stream>>>(
        ws + WS_RV, I_DIM, W_rd, b_rd, ws + WS_OUTH, out, I_DIM, O_DIM);
}